// GCNJointPredictor_6820408066337
// MI455X (gfx1250) — compile-verified
//
#include <hip/hip_runtime.h>
#include <math.h>

// ---------------------------------------------------------------------------
// GCN joint predictor forward for MI455X (gfx1250, wave32).
// Dense GEMMs use V_WMMA_F32_16X16X4_F32 (fp32 in/out, matches reference
// numerics).  Edge aggregation (the roofline-dominant part, ~6.6 GB of HBM
// traffic at 23.3 TB/s => ~280us floor) uses coalesced float4 gathers +
// global fp32 atomics.  GEMM fast path is fully unguarded (no EXEC churn
// around v_wmma); guarded path only runs on ragged edge tiles (574/152 cols,
// K=830 tail).
// ---------------------------------------------------------------------------

typedef __attribute__((ext_vector_type(2))) float v2f;
typedef __attribute__((ext_vector_type(8))) float v8f;

#define BN_INV 0.9999950000374997f  // 1/sqrt(1+1e-5)

// ---------------------------- utility kernels ------------------------------

__global__ void fill_u32_kernel(unsigned* __restrict__ p, unsigned v, long long n) {
    long long i = (long long)blockIdx.x * blockDim.x + threadIdx.x;
    if (i < n) p[i] = v;
}

__global__ void degree_kernel(const int* __restrict__ src, const int* __restrict__ dst,
                              int* __restrict__ degS, int* __restrict__ degD, int E) {
    int i = blockIdx.x * blockDim.x + threadIdx.x;
    if (i < E) {
        atomicAdd(&degS[src[i]], 1);
        atomicAdd(&degD[dst[i]], 1);
    }
}

__global__ void norm_kernel(const int* __restrict__ deg, float* __restrict__ nrm, int n) {
    int i = blockIdx.x * blockDim.x + threadIdx.x;
    if (i < n) nrm[i] = rsqrtf(fmaxf((float)deg[i], 1.0f));
}

// --------------------------- WMMA fp32 GEMM --------------------------------
// C[M x Nc] = act( (rowscale ? diag(rowscale) : I) * A[M x K] @ B[K x Nc] + bias )
// act: 0=none, 1=relu, 2=relu then BN (gamma * v * BN_INV + beta)
// One wave per 16x16 output tile; K stepped by 4 (V_WMMA_F32_16X16X4_F32).
// M must be a multiple of 16 (true for all call sites: 100000, 4096).
//
// fp32 A layout (16x4): lane l -> row l&15, VGPR0/1 = K pair (2*(l>>4), +1).
// fp32 B layout (4x16): lane l -> col l&15, VGPR0/1 = K pair (2*(l>>4), +1).
// fp32 C layout (16x16): lane l -> col l&15, VGPR r -> row r + 8*(l>>4).
__global__ void wmma_gemm_kernel(const float* __restrict__ A, const float* __restrict__ B,
                                 const float* __restrict__ bias,
                                 const float* __restrict__ rowscale,
                                 const float* __restrict__ gamma,
                                 const float* __restrict__ beta,
                                 float* __restrict__ C,
                                 int M, int Nc, int K, int act) {
    int tiles_n = (Nc + 15) >> 4;
    int tiles_m = M >> 4;
    int wave = blockIdx.x * (blockDim.x >> 5) + (threadIdx.x >> 5);
    if (wave >= tiles_m * tiles_n) return;      // wave-uniform: EXEC stays full
    int tm = wave / tiles_n;
    int tn = wave - tm * tiles_n;

    int lane = threadIdx.x & 31;
    int half = lane >> 4;                       // K-half select (0 or 1)
    int l15  = lane & 15;
    int arow = tm * 16 + l15;                   // A row for this lane
    int bcol = tn * 16 + l15;                   // B/C column for this lane

    float rs = rowscale ? rowscale[arow] : 1.0f;
    const float* Arow = A + (size_t)arow * K + 2 * half;   // lane's K pair base
    const int Kfull = K & ~3;

    v8f acc = {};

    if (tn * 16 + 16 <= Nc) {
        // ---------------- fast path: full column tile, no guards -----------
        const float* Bp = B + (size_t)2 * half * Nc + bcol; // lane's K pair base
        #pragma unroll 4
        for (int k0 = 0; k0 < Kfull; k0 += 4) {
            v2f a = *(const v2f*)(Arow + k0);               // global_load_b64
            a.x *= rs;
            a.y *= rs;
            v2f b;
            b.x = Bp[(size_t)k0 * Nc];
            b.y = Bp[(size_t)k0 * Nc + Nc];
            acc = __builtin_amdgcn_wmma_f32_16x16x4_f32(
                false, a, false, b, (short)0, acc, false, false);
        }
        if (Kfull < K) {                                    // K tail (K=830)
            int ka = Kfull + 2 * half;
            v2f a, b;
            a.x = (ka     < K) ? Arow[Kfull]     * rs : 0.0f;
            a.y = (ka + 1 < K) ? Arow[Kfull + 1] * rs : 0.0f;
            b.x = (ka     < K) ? B[(size_t)ka       * Nc + bcol] : 0.0f;
            b.y = (ka + 1 < K) ? B[(size_t)(ka + 1) * Nc + bcol] : 0.0f;
            acc = __builtin_amdgcn_wmma_f32_16x16x4_f32(
                false, a, false, b, (short)0, acc, false, false);
        }
        // epilogue (unguarded stores)
        float bi = bias  ? bias[bcol]  : 0.0f;
        float ga = gamma ? gamma[bcol] : 1.0f;
        float be = beta  ? beta[bcol]  : 0.0f;
        #pragma unroll
        for (int r = 0; r < 8; ++r) {
            int orow = tm * 16 + r + 8 * half;
            float v = acc[r] + bi;
            if (act >= 1) v = fmaxf(v, 0.0f);
            if (act == 2) v = ga * (v * BN_INV) + be;
            C[(size_t)orow * Nc + bcol] = v;
        }
    } else {
        // -------------- guarded path: ragged last column tile --------------
        bool colok = (bcol < Nc);
        for (int k0 = 0; k0 < K; k0 += 4) {
            int ka = k0 + 2 * half;
            v2f a, b;
            a.x = (ka     < K) ? Arow[k0]     * rs : 0.0f;
            a.y = (ka + 1 < K) ? Arow[k0 + 1] * rs : 0.0f;
            b.x = (colok && ka     < K) ? B[(size_t)ka       * Nc + bcol] : 0.0f;
            b.y = (colok && ka + 1 < K) ? B[(size_t)(ka + 1) * Nc + bcol] : 0.0f;
            acc = __builtin_amdgcn_wmma_f32_16x16x4_f32(
                false, a, false, b, (short)0, acc, false, false);
        }
        if (colok) {
            float bi = bias  ? bias[bcol]  : 0.0f;
            float ga = gamma ? gamma[bcol] : 1.0f;
            float be = beta  ? beta[bcol]  : 0.0f;
            #pragma unroll
            for (int r = 0; r < 8; ++r) {
                int orow = tm * 16 + r + 8 * half;
                float v = acc[r] + bi;
                if (act >= 1) v = fmaxf(v, 0.0f);
                if (act == 2) v = ga * (v * BN_INV) + be;
                C[(size_t)orow * Nc + bcol] = v;
            }
        }
    }
}

// -------------------------- edge scatter (agg) -----------------------------
// One wave per edge: 32 lanes x float4 = 128-float row.  agg[dst] += h[src].
__global__ void edge_scatter_kernel(const float* __restrict__ h,
                                    const int* __restrict__ src,
                                    const int* __restrict__ dst,
                                    float* __restrict__ agg, int E) {
    int e = blockIdx.x * (blockDim.x >> 5) + (threadIdx.x >> 5);
    if (e >= E) return;
    int lane = threadIdx.x & 31;
    int s = src[e];
    int d = dst[e];
    const float4 v = *(const float4*)(h + (size_t)s * 128 + lane * 4);
    float* out = agg + (size_t)d * 128 + lane * 4;
    atomicAdd(out + 0, v.x);
    atomicAdd(out + 1, v.y);
    atomicAdd(out + 2, v.z);
    atomicAdd(out + 3, v.w);
}

// out = relu(agg * in_norm[row] + b[col]) + res    (res already relu'd)
__global__ void combine_kernel(const float* __restrict__ agg, const float* __restrict__ res,
                               const float* __restrict__ in_norm, const float* __restrict__ b,
                               float* __restrict__ out, long long total) {
    long long i = (long long)blockIdx.x * blockDim.x + threadIdx.x;
    if (i < total) {
        int row = (int)(i >> 7);
        int col = (int)(i & 127);
        out[i] = fmaxf(agg[i] * in_norm[row] + b[col], 0.0f) + res[i];
    }
}

// ------------------------------- readout -----------------------------------

__device__ __forceinline__ void atomic_max_f32(float* addr, float val) {
    if (val >= 0.0f)
        atomicMax((int*)addr, __float_as_int(val));
    else
        atomicMin((unsigned int*)addr, __float_as_uint(val));
}

// One wave per node: aw = sigmoid(h . aw_W + aw_b);
// hsum[g] += aw*h (atomic add), hmax[g] = max (atomic max).
__global__ void readout_kernel(const float* __restrict__ h, const float* __restrict__ awW,
                               const float* __restrict__ awb, const int* __restrict__ gid,
                               float* __restrict__ hsum, float* __restrict__ hmax, int N) {
    int node = blockIdx.x * (blockDim.x >> 5) + (threadIdx.x >> 5);
    if (node >= N) return;
    int lane = threadIdx.x & 31;
    const float4 hv = *(const float4*)(h + (size_t)node * 128 + lane * 4);
    const float4 wv = *(const float4*)(awW + lane * 4);
    float p = hv.x * wv.x + hv.y * wv.y + hv.z * wv.z + hv.w * wv.w;
    #pragma unroll
    for (int off = 16; off > 0; off >>= 1) p += __shfl_xor(p, off, 32);
    float aw = 1.0f / (1.0f + __expf(-(p + awb[0])));
    int g = gid[node];
    float* hs = hsum + (size_t)g * 128 + lane * 4;
    atomicAdd(hs + 0, aw * hv.x);
    atomicAdd(hs + 1, aw * hv.y);
    atomicAdd(hs + 2, aw * hv.z);
    atomicAdd(hs + 3, aw * hv.w);
    float* hm = hmax + (size_t)g * 128 + lane * 4;
    atomic_max_f32(hm + 0, hv.x);
    atomic_max_f32(hm + 1, hv.y);
    atomic_max_f32(hm + 2, hv.z);
    atomic_max_f32(hm + 3, hv.w);
}

// g = [hsum | guard(hmax)]  (B x 256)
__global__ void g_assemble_kernel(const float* __restrict__ hsum, const float* __restrict__ hmax,
                                  float* __restrict__ g, long long total) {
    long long i = (long long)blockIdx.x * blockDim.x + threadIdx.x;
    if (i < total) {
        int row = (int)(i >> 8);
        int col = (int)(i & 255);
        if (col < 128) {
            g[i] = hsum[(size_t)row * 128 + col];
        } else {
            float v = hmax[(size_t)row * 128 + (col - 128)];
            g[i] = (v >= -3.0e38f && v <= 3.0e38f) ? v : 0.0f;  // -inf guard
        }
    }
}

// x2 = [g | or_logits]  (B x 830)
__global__ void x2_assemble_kernel(const float* __restrict__ g, const float* __restrict__ orlog,
                                   float* __restrict__ x2, long long total) {
    long long i = (long long)blockIdx.x * blockDim.x + threadIdx.x;
    if (i < total) {
        int row = (int)(i / 830);
        int col = (int)(i - (long long)row * 830);
        x2[i] = (col < 256) ? g[(size_t)row * 256 + col]
                            : orlog[(size_t)row * 574 + (col - 256)];
    }
}

// ------------------------------- launcher ----------------------------------

extern "C" void kernel_launch(void* const* d_in, const int* in_sizes, int n_in,
                              void* d_out, int out_size, void* d_ws, size_t ws_size,
                              hipStream_t stream) {
    const float* feats = (const float*)d_in[0];
    const int*   src   = (const int*)d_in[1];
    const int*   dst   = (const int*)d_in[2];
    const int*   gid   = (const int*)d_in[3];
    const float* W1    = (const float*)d_in[4];
    const float* b1    = (const float*)d_in[5];
    const float* resW1 = (const float*)d_in[6];
    const float* resb1 = (const float*)d_in[7];
    const float* W2    = (const float*)d_in[8];
    const float* b2    = (const float*)d_in[9];
    const float* resW2 = (const float*)d_in[10];
    const float* resb2 = (const float*)d_in[11];
    const float* awW   = (const float*)d_in[12];
    const float* awb   = (const float*)d_in[13];
    const float* orW1  = (const float*)d_in[14];
    const float* orb1  = (const float*)d_in[15];
    const float* org   = (const float*)d_in[16];
    const float* orbeta= (const float*)d_in[17];
    const float* orW2  = (const float*)d_in[18];
    const float* orb2  = (const float*)d_in[19];
    const float* scW1  = (const float*)d_in[20];
    const float* scb1  = (const float*)d_in[21];
    const float* scg   = (const float*)d_in[22];
    const float* scbeta= (const float*)d_in[23];
    const float* scW2  = (const float*)d_in[24];
    const float* scb2  = (const float*)d_in[25];

    const int N  = in_sizes[0] / 128;    // 100000
    const int E  = in_sizes[1];          // 3200000
    const int T1 = 574, T2 = 152;
    const int B  = out_size / (T1 + T2); // 4096

    // ---- workspace layout (256B aligned slabs) ----
    char* ws = (char*)d_ws;
    size_t off = 0;
    auto salloc = [&](size_t bytes) -> void* {
        void* p = ws + off;
        off = (off + bytes + 255) & ~(size_t)255;
        return p;
    };
    int*   degS   = (int*)  salloc((size_t)N * 4);
    int*   degD   = (int*)  salloc((size_t)N * 4);
    float* onorm  = (float*)salloc((size_t)N * 4);
    float* inorm  = (float*)salloc((size_t)N * 4);
    float* Htmp   = (float*)salloc((size_t)N * 128 * 4);
    float* Agg    = (float*)salloc((size_t)N * 128 * 4);
    float* Hcur   = (float*)salloc((size_t)N * 128 * 4);
    float* hsum   = (float*)salloc((size_t)B * 128 * 4);
    float* hmax   = (float*)salloc((size_t)B * 128 * 4);
    float* gbuf   = (float*)salloc((size_t)B * 256 * 4);
    float* zbuf   = (float*)salloc((size_t)B * 128 * 4);
    float* x2buf  = (float*)salloc((size_t)B * 830 * 4);
    (void)ws_size; (void)n_in;

    const int TPB = 256;                 // 8 waves / block
    auto gblk = [&](long long n) { return (int)((n + TPB - 1) / TPB); };

    auto gemm = [&](const float* A, const float* Bm, const float* bias,
                    const float* rowscale, const float* gamma, const float* beta,
                    float* C, int M, int Nc, int K, int act) {
        int tiles = (M >> 4) * ((Nc + 15) >> 4);
        int blocks = (tiles + 7) / 8;    // 8 waves per block
        wmma_gemm_kernel<<<blocks, TPB, 0, stream>>>(A, Bm, bias, rowscale,
                                                     gamma, beta, C, M, Nc, K, act);
    };

    // ---- degrees and norms ----
    fill_u32_kernel<<<gblk(N), TPB, 0, stream>>>((unsigned*)degS, 0u, N);
    fill_u32_kernel<<<gblk(N), TPB, 0, stream>>>((unsigned*)degD, 0u, N);
    degree_kernel<<<gblk(E), TPB, 0, stream>>>(src, dst, degS, degD, E);
    norm_kernel<<<gblk(N), TPB, 0, stream>>>(degS, onorm, N);
    norm_kernel<<<gblk(N), TPB, 0, stream>>>(degD, inorm, N);

    const long long NH = (long long)N * 128;

    // ---- GCN layer 1 ----
    gemm(feats, W1, nullptr, onorm, nullptr, nullptr, Htmp, N, 128, 128, 0);
    fill_u32_kernel<<<gblk(NH), TPB, 0, stream>>>((unsigned*)Agg, 0u, NH);
    edge_scatter_kernel<<<(E + 7) / 8, TPB, 0, stream>>>(Htmp, src, dst, Agg, E);
    gemm(feats, resW1, resb1, nullptr, nullptr, nullptr, Htmp, N, 128, 128, 1);
    combine_kernel<<<gblk(NH), TPB, 0, stream>>>(Agg, Htmp, inorm, b1, Hcur, NH);

    // ---- GCN layer 2 ----
    gemm(Hcur, W2, nullptr, onorm, nullptr, nullptr, Htmp, N, 128, 128, 0);
    fill_u32_kernel<<<gblk(NH), TPB, 0, stream>>>((unsigned*)Agg, 0u, NH);
    edge_scatter_kernel<<<(E + 7) / 8, TPB, 0, stream>>>(Htmp, src, dst, Agg, E);
    gemm(Hcur, resW2, resb2, nullptr, nullptr, nullptr, Htmp, N, 128, 128, 1);
    combine_kernel<<<gblk(NH), TPB, 0, stream>>>(Agg, Htmp, inorm, b2, Hcur, NH);

    // ---- WeightedSumAndMax readout ----
    const long long BH = (long long)B * 128;
    fill_u32_kernel<<<gblk(BH), TPB, 0, stream>>>((unsigned*)hsum, 0u, BH);
    fill_u32_kernel<<<gblk(BH), TPB, 0, stream>>>((unsigned*)hmax, 0xFF800000u, BH); // -inf
    readout_kernel<<<(N + 7) / 8, TPB, 0, stream>>>(Hcur, awW, awb, gid, hsum, hmax, N);
    g_assemble_kernel<<<gblk((long long)B * 256), TPB, 0, stream>>>(hsum, hmax, gbuf,
                                                                    (long long)B * 256);

    // ---- MLP head 1: or_logits ----
    float* orlog = (float*)d_out;                       // [B, 574]
    gemm(gbuf, orW1, orb1, nullptr, org, orbeta, zbuf, B, 128, 256, 2);
    gemm(zbuf, orW2, orb2, nullptr, nullptr, nullptr, orlog, B, T1, 128, 0);

    // ---- MLP head 2: scent ----
    x2_assemble_kernel<<<gblk((long long)B * 830), TPB, 0, stream>>>(gbuf, orlog, x2buf,
                                                                     (long long)B * 830);
    gemm(x2buf, scW1, scb1, nullptr, scg, scbeta, zbuf, B, 128, 830, 2);
    gemm(zbuf, scW2, scb2, nullptr, nullptr, nullptr, orlog + (size_t)B * T1, B, T2, 128, 0);
}